// RGCN_14053132992552
// MI455X (gfx1250) — compile-verified
//
#include <hip/hip_runtime.h>

// ---------------------------------------------------------------------------
// Problem constants (from the reference)
// ---------------------------------------------------------------------------
#define N_ENT   100000
#define N_REL   8
#define DIM     128
#define N_BASES 4
#define N_EDGES 1600000
#define N_TRIP  8192

#define K_EXT   ((N_REL + 1) * DIM)      // 1152: 8 relation slots + root slot
#define KC_TOT  (K_EXT / 4)              // 288 WMMA K-steps
#define M_TILES (N_ENT / 16)             // 6250 (exact)

typedef float v2f __attribute__((ext_vector_type(2)));
typedef float v8f __attribute__((ext_vector_type(8)));

// ---------------------------------------------------------------------------
// 0) grid-stride float4 zero
// ---------------------------------------------------------------------------
__global__ void zero_f32(float* __restrict__ p, long long n4) {
    long long i = (long long)blockIdx.x * blockDim.x + threadIdx.x;
    long long stride = (long long)gridDim.x * blockDim.x;
    float4* p4 = (float4*)p;
    float4 z = make_float4(0.f, 0.f, 0.f, 0.f);
    for (; i < n4; i += stride) p4[i] = z;
}

__global__ void zero_one(float* p) { *p = 0.f; }

// ---------------------------------------------------------------------------
// 1) edge counts per (dst, rel) bucket — kept as f32 so 1/cnt is free
// ---------------------------------------------------------------------------
__global__ void count_edges(const int* __restrict__ dst,
                            const int* __restrict__ etype,
                            float* __restrict__ cnt) {
    int e = blockIdx.x * blockDim.x + threadIdx.x;
    if (e < N_EDGES)
        unsafeAtomicAdd(&cnt[(size_t)dst[e] * N_REL + etype[e]], 1.0f);
}

// ---------------------------------------------------------------------------
// 2) build W_ext = [einsum(comp,basis) ; root] packed into WMMA B-fragment
//    order.  B frag for K-step kc, N-tile nt: lane holds (W[k][n], W[k+1][n])
//    with k = 4*kc + (lane<16 ? 0 : 2), n = 16*nt + (lane&15).
// ---------------------------------------------------------------------------
__global__ void build_wpack(const float* __restrict__ comp,   // [R, B]
                            const float* __restrict__ basis,  // [B, D, D]
                            const float* __restrict__ root,   // [D, D]
                            float* __restrict__ wpack) {
    int idx = blockIdx.x * blockDim.x + threadIdx.x;   // over K_EXT * DIM
    if (idx >= K_EXT * DIM) return;
    int k = idx >> 7;         // 0..1151
    int o = idx & 127;
    float w;
    if (k < N_REL * DIM) {
        int r = k >> 7;
        int i = k & 127;
        const float* c = comp + r * N_BASES;
        w = c[0] * basis[0 * DIM * DIM + i * DIM + o]
          + c[1] * basis[1 * DIM * DIM + i * DIM + o]
          + c[2] * basis[2 * DIM * DIM + i * DIM + o]
          + c[3] * basis[3 * DIM * DIM + i * DIM + o];
    } else {
        w = root[(k - N_REL * DIM) * DIM + o];
    }
    int kc   = k >> 2;
    int koff = k & 3;
    int lane = ((koff >> 1) << 4) | (o & 15);
    int nt   = o >> 4;
    wpack[(((size_t)kc * 8 + nt) * 32 + lane) * 2 + (koff & 1)] = w;
}

// ---------------------------------------------------------------------------
// 3) scatter: one wave per edge, lane = 4 floats of the 128-wide feature row.
//    Mean is folded in by scaling each contribution with 1/cnt[dst,rel].
// ---------------------------------------------------------------------------
__global__ void scatter_edges(const int* __restrict__ src,
                              const int* __restrict__ dst,
                              const int* __restrict__ etype,
                              const float* __restrict__ cnt,
                              const float* __restrict__ X,      // [N, D]
                              float* __restrict__ agg) {        // [N, R, D]
    int e = (blockIdx.x * blockDim.x + threadIdx.x) >> 5;   // wave id (uniform)
    if (e >= N_EDGES) return;
    int lane = threadIdx.x & 31;
    int s = src[e];
    size_t seg = (size_t)dst[e] * N_REL + etype[e];
    float inv = 1.0f / cnt[seg];                 // cnt >= 1 for this bucket
    const float4 v = ((const float4*)(X + (size_t)s * DIM))[lane];
    float* dp = agg + seg * DIM + lane * 4;
    unsafeAtomicAdd(dp + 0, v.x * inv);
    unsafeAtomicAdd(dp + 1, v.y * inv);
    unsafeAtomicAdd(dp + 2, v.z * inv);
    unsafeAtomicAdd(dp + 3, v.w * inv);
}

// ---------------------------------------------------------------------------
// 4) WMMA GEMM: Y[M=100000, N=128] = A_ext[M, 1152] @ W_ext[1152, 128] (+bias)
//    A_ext row n = [agg[n, 0..7, :], X[n, :]].  One wave per 16-row stripe,
//    full N held in 8 v8f accumulators, K looped in steps of 4 with
//    v_wmma_f32_16x16x4_f32 (exact f32).  EXEC stays all-ones per wave.
// ---------------------------------------------------------------------------
__global__ void __launch_bounds__(256)
rgcn_gemm(const float* __restrict__ agg,    // [N, R*D]
          const float* __restrict__ X,      // [N, D]
          const float* __restrict__ wpack,  // packed B fragments
          const float* __restrict__ bias,   // [D]
          float* __restrict__ Y,            // [N, D]
          int do_relu) {
    const int lane  = threadIdx.x & 31;
    const int wid   = threadIdx.x >> 5;
    const int mTile = blockIdx.x * 8 + wid;
    if (mTile >= M_TILES) return;                 // wave-uniform guard
    const int m0 = mTile * 16;

    // f32 WMMA A-frag: lanes 0-15 -> K=k0,k0+1 ; lanes 16-31 -> K=k0+2,k0+3
    const int mrow  = m0 + (lane & 15);
    const int koff2 = (lane >> 4) << 1;           // 0 or 2
    const float* arow = agg + (size_t)mrow * (N_REL * DIM) + koff2;
    const float* xrow = X   + (size_t)mrow * DIM + koff2;

    v8f acc[8] = {};

    // K = 0..1023 : relation-aggregated features
    #pragma unroll 4
    for (int kc = 0; kc < (N_REL * DIM) / 4; ++kc) {
        v2f a = *(const v2f*)(arow + kc * 4);
        #pragma unroll
        for (int nt = 0; nt < 8; ++nt) {
            v2f b = *(const v2f*)(wpack + (((size_t)kc * 8 + nt) * 32 + lane) * 2);
            acc[nt] = __builtin_amdgcn_wmma_f32_16x16x4_f32(
                false, a, false, b, (short)0, acc[nt], false, false);
        }
    }
    // K = 1024..1151 : self (root) term
    #pragma unroll 4
    for (int kc = (N_REL * DIM) / 4; kc < KC_TOT; ++kc) {
        v2f a = *(const v2f*)(xrow + (kc - (N_REL * DIM) / 4) * 4);
        #pragma unroll
        for (int nt = 0; nt < 8; ++nt) {
            v2f b = *(const v2f*)(wpack + (((size_t)kc * 8 + nt) * 32 + lane) * 2);
            acc[nt] = __builtin_amdgcn_wmma_f32_16x16x4_f32(
                false, a, false, b, (short)0, acc[nt], false, false);
        }
    }

    // C/D layout: VGPR v -> row m0+v (lanes 0-15) / m0+v+8 (lanes 16-31)
    const int rbase = m0 + ((lane >> 4) << 3);
    const int cbase = lane & 15;
    #pragma unroll
    for (int v = 0; v < 8; ++v) {
        int row = rbase + v;
        #pragma unroll
        for (int nt = 0; nt < 8; ++nt) {
            int col = nt * 16 + cbase;
            float val = acc[nt][v] + bias[col];
            if (do_relu) val = fmaxf(val, 0.0f);
            Y[(size_t)row * DIM + col] = val;
        }
    }
}

// ---------------------------------------------------------------------------
// 5) DistMult + BCE-with-logits loss.  One wave per triplet.
// ---------------------------------------------------------------------------
__device__ __forceinline__ float softplus_stable(float z) {
    return fmaxf(z, 0.0f) + log1pf(expf(-fabsf(z)));
}

__global__ void distmult_loss(const int* __restrict__ trip,     // [T,3]
                              const float* __restrict__ labels, // [T]
                              const float* __restrict__ ns,     // node_state
                              const float* __restrict__ rel_emb,// [R, D]
                              float* __restrict__ loss) {
    int t = (blockIdx.x * blockDim.x + threadIdx.x) >> 5;
    if (t >= N_TRIP) return;
    int lane = threadIdx.x & 31;
    int h = trip[t * 3 + 0];
    int r = trip[t * 3 + 1];
    int a = trip[t * 3 + 2];
    const float4 hv = ((const float4*)(ns + (size_t)h * DIM))[lane];
    const float4 rv = ((const float4*)(rel_emb + (size_t)r * DIM))[lane];
    const float4 av = ((const float4*)(ns + (size_t)a * DIM))[lane];
    float p = hv.x * rv.x * av.x + hv.y * rv.y * av.y
            + hv.z * rv.z * av.z + hv.w * rv.w * av.w;
    #pragma unroll
    for (int off = 16; off > 0; off >>= 1)
        p += __shfl_xor(p, off, 32);
    if (lane == 0) {
        float lbl = labels[t];
        // -lbl*logsig(z) - (1-lbl)*logsig(-z) = lbl*sp(-z) + (1-lbl)*sp(z)
        float term = lbl * softplus_stable(-p) + (1.0f - lbl) * softplus_stable(p);
        unsafeAtomicAdd(loss, term * (1.0f / (float)N_TRIP));
    }
}

// ---------------------------------------------------------------------------
// Launch
// ---------------------------------------------------------------------------
extern "C" void kernel_launch(void* const* d_in, const int* in_sizes, int n_in,
                              void* d_out, int out_size, void* d_ws, size_t ws_size,
                              hipStream_t stream) {
    // inputs (setup_inputs order)
    const int*   edge_index = (const int*)d_in[1];       // [2, E]
    const int*   src        = edge_index;
    const int*   dst        = edge_index + N_EDGES;
    const int*   etype      = (const int*)d_in[2];
    const int*   trip       = (const int*)d_in[3];
    const float* labels     = (const float*)d_in[4];
    const float* ent_embed  = (const float*)d_in[5];
    const float* rel_embed  = (const float*)d_in[6];
    const float* comp1      = (const float*)d_in[7];
    const float* basis1     = (const float*)d_in[8];
    const float* root1      = (const float*)d_in[9];
    const float* bias1      = (const float*)d_in[10];
    const float* comp2      = (const float*)d_in[11];
    const float* basis2     = (const float*)d_in[12];
    const float* root2      = (const float*)d_in[13];
    const float* bias2      = (const float*)d_in[14];

    float* node_state = (float*)d_out;                       // [N, D]
    float* loss       = node_state + (size_t)N_ENT * DIM;    // 1 scalar

    // workspace layout (floats)
    float* agg   = (float*)d_ws;                                        // 102.4M
    float* cnt   = agg + (size_t)N_ENT * N_REL * DIM;                   // 800K
    float* h     = cnt + (size_t)N_ENT * N_REL;                         // 12.8M
    float* wpack = h + (size_t)N_ENT * DIM;                             // 147456

    const long long AGG_F  = (long long)N_ENT * N_REL * DIM;
    const long long CNT_F  = (long long)N_ENT * N_REL;

    dim3 blk(256);

    // ---- counts (reused by both layers) + zero agg -------------------------
    zero_f32<<<2048, blk, 0, stream>>>(agg, (AGG_F + CNT_F) / 4);
    count_edges<<<(N_EDGES + 255) / 256, blk, 0, stream>>>(dst, etype, cnt);

    // ---- layer 1: X = entity_embed -> h (ReLU) -----------------------------
    build_wpack<<<(K_EXT * DIM + 255) / 256, blk, 0, stream>>>(comp1, basis1, root1, wpack);
    scatter_edges<<<(N_EDGES * 32 + 255) / 256, blk, 0, stream>>>(src, dst, etype, cnt, ent_embed, agg);
    rgcn_gemm<<<(M_TILES + 7) / 8, blk, 0, stream>>>(agg, ent_embed, wpack, bias1, h, 1);

    // ---- layer 2: X = h -> node_state (no ReLU) ----------------------------
    zero_f32<<<2048, blk, 0, stream>>>(agg, AGG_F / 4);
    build_wpack<<<(K_EXT * DIM + 255) / 256, blk, 0, stream>>>(comp2, basis2, root2, wpack);
    scatter_edges<<<(N_EDGES * 32 + 255) / 256, blk, 0, stream>>>(src, dst, etype, cnt, h, agg);
    rgcn_gemm<<<(M_TILES + 7) / 8, blk, 0, stream>>>(agg, h, wpack, bias2, node_state, 0);

    // ---- DistMult loss -----------------------------------------------------
    zero_one<<<1, 1, 0, stream>>>(loss);
    distmult_loss<<<(N_TRIP * 32 + 255) / 256, blk, 0, stream>>>(trip, labels, node_state, rel_embed, loss);
}